// GreedyGenerator_54597624266912
// MI455X (gfx1250) — compile-verified
//
#include <hip/hip_runtime.h>

#define BEAM   5
#define NHYP   10
#define VOCAB  768
#define TSEQ   180
#define SMAX   180
#define NSTEPS 179              // min(S, T) - 1
#define PAD    0
#define BOS    1
#define EOS    2
#define NCAND  (BEAM * VOCAB)   // 3840 = 15 * 256 candidates per step
#define NCHUNK (NCAND / 4)      // 960 16-byte chunks per step tile
#define BLOCK  256
#define NWAVE  (BLOCK / 32)

// ---- CDNA5 async global->LDS copy (ASYNCcnt-tracked), GVS addressing ----
__device__ __forceinline__ void async_load_b128(unsigned lds_addr,
                                                const void* gbase,
                                                int goff_bytes) {
  // vdst = LDS byte address, vaddr = signed 32-bit byte offset, saddr = 64-bit base
  asm volatile("global_load_async_to_lds_b128 %0, %1, %2"
               :
               : "v"(lds_addr), "v"(goff_bytes), "s"(gbase)
               : "memory");
}

__device__ __forceinline__ void wait_async_le4() {
  // each wave issues exactly 4 async loads per step; async loads complete in
  // order, so <=4 outstanding guarantees the previous step's tile is resident
  asm volatile("s_wait_asynccnt 4" ::: "memory");
}

__device__ __forceinline__ void wait_async0() {
  asm volatile("s_wait_asynccnt 0" ::: "memory");
}

__device__ __forceinline__ unsigned lds_off(const void* p) {
  // flat LDS address: low 32 bits are the wave-relative LDS byte address
  return (unsigned)(unsigned long long)p;
}

__global__ __launch_bounds__(BLOCK)
void beam_search_kernel(const float* __restrict__ all_scores,
                        int* __restrict__ out_seqs,     // [B*BEAM][SMAX]
                        int* __restrict__ out_lens,     // [B*BEAM]
                        float* __restrict__ out_scores) // [B*BEAM]
{
  __shared__ float sc[2][NCAND];          // double-buffered step tile (async target)
  __shared__ int   gen[2][BEAM][SMAX];    // double-buffered generated tokens
  __shared__ int   hyp_tok[NHYP][SMAX];
  __shared__ float hyp_score[NHYP];
  __shared__ float beam_sc[BEAM];
  __shared__ float live_sc[BEAM];
  __shared__ int   live_w[BEAM];
  __shared__ int   live_e[BEAM];
  __shared__ float topv[NHYP];
  __shared__ int   topi[NHYP];
  __shared__ float wredv[NWAVE];
  __shared__ int   wredi[NWAVE];
  __shared__ int   win_idx;
  __shared__ int   ins_slot[NHYP];
  __shared__ int   ins_src[NHYP];
  __shared__ int   st_done;
  __shared__ int   st_cnt;
  __shared__ int   order[NHYP];

  const int tid  = threadIdx.x;
  const int lane = tid & 31;
  const int wv_id = tid >> 5;
  const int b    = blockIdx.x;
  const float* batch_base = all_scores + (size_t)b * BEAM * TSEQ * VOCAB;

  // ---- init persistent LDS state ----
  for (int i = tid; i < BEAM * SMAX; i += BLOCK)
    gen[0][i / SMAX][i % SMAX] = ((i % SMAX) == 0) ? BOS : PAD;
  for (int i = tid; i < NHYP * SMAX; i += BLOCK)
    hyp_tok[i / SMAX][i % SMAX] = PAD;
  if (tid < NHYP) hyp_score[tid] = -1e18f;
  if (tid < BEAM) beam_sc[tid] = (tid == 0) ? 0.0f : -1e9f;
  if (tid == 0) { st_done = 0; st_cnt = 0; }
  __syncthreads();

  // ---- preload step t=1 tile into buffer 1: exactly 4 async loads per wave ----
  for (int k4 = 0; k4 < 4; ++k4) {
    int k = tid + (k4 << 8);
    if (k >= NCHUNK) k = NCHUNK - 1;          // duplicate tail chunk (benign)
    const int bm   = k / (VOCAB / 4);
    const int off  = (k % (VOCAB / 4)) * 16;
    const int goff = (bm * TSEQ + 1) * VOCAB * 4 + off;
    async_load_b128(lds_off(&sc[1][k * 4]), (const void*)batch_base, goff);
  }

  int genbuf = 0;
  for (int t = 1; t <= NSTEPS; ++t) {
    const int cur = t & 1;

    // ---- issue prefetch of step t+1 FIRST, then wait for current tile ----
    if (t < NSTEPS) {
      const int nb = (t + 1) & 1;
      for (int k4 = 0; k4 < 4; ++k4) {
        int k = tid + (k4 << 8);
        if (k >= NCHUNK) k = NCHUNK - 1;
        const int bm   = k / (VOCAB / 4);
        const int off  = (k % (VOCAB / 4)) * 16;
        const int goff = (bm * TSEQ + (t + 1)) * VOCAB * 4 + off;
        async_load_b128(lds_off(&sc[nb][k * 4]), (const void*)batch_base, goff);
      }
      wait_async_le4();                      // current tile done, prefetch in flight
    } else {
      wait_async0();
    }
    __syncthreads();                         // publish current tile to all waves

    // ---- top-10: lazy per-thread local max + wave32 butterfly reduce ----
    // each thread owns candidates {tid + 256k : k=0..14}
    float bv = -3.0e38f; int bi = 0;
    #pragma unroll
    for (int k = 0; k < NCAND / BLOCK; ++k) {
      const int i = tid + (k << 8);
      const float v = sc[cur][i] + beam_sc[i / VOCAB];
      if (v > bv) { bv = v; bi = i; }
    }

    for (int r = 0; r < NHYP; ++r) {
      // wave32 butterfly max (value desc, index asc on ties) — no barriers
      float mv = bv; int mi = bi;
      #pragma unroll
      for (int off = 16; off > 0; off >>= 1) {
        const float ov = __shfl_xor(mv, off, 32);
        const int   oi = __shfl_xor(mi, off, 32);
        if (ov > mv || (ov == mv && oi < mi)) { mv = ov; mi = oi; }
      }
      if (lane == 0) { wredv[wv_id] = mv; wredi[wv_id] = mi; }
      __syncthreads();
      if (tid == 0) {
        float gv = wredv[0]; int gi = wredi[0];
        for (int w = 1; w < NWAVE; ++w)
          if (wredv[w] > gv || (wredv[w] == gv && wredi[w] < gi)) { gv = wredv[w]; gi = wredi[w]; }
        topv[r] = gv; topi[r] = gi;
        sc[cur][gi] = -3.0e38f;              // mask winner
        win_idx = gi;
      }
      __syncthreads();
      if (r + 1 < NHYP) {
        // only the owner of the masked index rescans its 15 elements
        if ((win_idx & (BLOCK - 1)) == tid) {
          bv = -3.0e38f; bi = 0;
          #pragma unroll
          for (int k = 0; k < NCAND / BLOCK; ++k) {
            const int i = tid + (k << 8);
            const float v = sc[cur][i] + beam_sc[i / VOCAB];
            if (v > bv) { bv = v; bi = i; }
          }
        }
      }
    }

    // ---- serial hypothesis bookkeeping (thread 0) ----
    if (tid == 0) {
      float minhyp = hyp_score[0];
      for (int i = 1; i < NHYP; ++i) minhyp = fminf(minhyp, hyp_score[i]);
      int done = st_done | ((st_cnt >= NHYP) && (minhyp >= topv[0]));
      st_done = done;

      int n_live = 0;
      const int is_last = (t + 1 == SMAX);
      for (int c = 0; c < NHYP; ++c) {
        ins_slot[c] = -1;
        const int   idx     = topi[c];
        const float value   = topv[c];
        const int   beam_id = idx / VOCAB;
        const int   word_id = idx % VOCAB;
        const int   is_eos  = (word_id == EOS) || is_last;
        const int   active  = (!done) && (n_live < BEAM);
        if (active && is_eos) {
          const float score_n = value / (float)t;
          float worst = hyp_score[0]; int amin = 0;
          for (int i = 1; i < NHYP; ++i)
            if (hyp_score[i] < worst) { worst = hyp_score[i]; amin = i; }
          const int insert = (st_cnt < NHYP) || (score_n > worst);
          const int slot   = (st_cnt < NHYP) ? st_cnt : amin;
          if (insert) {
            hyp_score[slot] = score_n;
            ins_slot[c] = slot;
            ins_src[c]  = beam_id;
            st_cnt = min(st_cnt + 1, NHYP);
          }
        } else if (active) {
          live_sc[n_live] = value;
          live_w[n_live]  = word_id;
          live_e[n_live]  = beam_id;
          ++n_live;
        }
      }
      for (int j = n_live; j < BEAM; ++j) { live_sc[j] = 0.0f; live_w[j] = PAD; live_e[j] = 0; }
    }
    __syncthreads();

    // ---- parallel: capture finished hypotheses (reads OLD gen buffer).
    // Token position p is owned by thread p%BLOCK, which walks c in order,
    // preserving sequential-overwrite semantics for repeated slots.
    for (int c = 0; c < NHYP; ++c) {
      const int slot = ins_slot[c];
      if (slot >= 0) {
        const int src = ins_src[c];
        for (int p = tid; p < SMAX; p += BLOCK)
          hyp_tok[slot][p] = gen[genbuf][src][p];
      }
    }

    // ---- parallel: beam permutation gen_new = gen_old[live_eff], [t] = live_word ----
    {
      const int nb2 = genbuf ^ 1;
      for (int i = tid; i < BEAM * SMAX; i += BLOCK) {
        const int j = i / SMAX, p = i % SMAX;
        int v = gen[genbuf][live_e[j]][p];
        if (p == t) v = live_w[j];
        gen[nb2][j][p] = v;
      }
    }
    if (tid < BEAM) beam_sc[tid] = live_sc[tid];
    __syncthreads();
    genbuf ^= 1;
  }

  // ---- final: stable argsort(-hyp_score), take top BEAM, emit ----
  if (tid == 0) {
    float s[NHYP]; int id[NHYP];
    for (int i = 0; i < NHYP; ++i) { s[i] = hyp_score[i]; id[i] = i; }
    for (int i = 0; i < NHYP; ++i) {
      int m = i;
      for (int j = i + 1; j < NHYP; ++j)
        if (s[j] > s[m] || (s[j] == s[m] && id[j] < id[m])) m = j;
      float tv = s[m]; s[m] = s[i]; s[i] = tv;
      int   ti = id[m]; id[m] = id[i]; id[i] = ti;
      order[i] = id[i];
    }
  }
  __syncthreads();

  const int cnt = st_cnt;
  for (int j = 0; j < BEAM; ++j) {
    const int valid = (j < cnt);
    const int row   = b * BEAM + j;
    const int src   = order[j];
    for (int p = tid; p < SMAX; p += BLOCK)
      out_seqs[row * SMAX + p] = valid ? hyp_tok[src][p] : PAD;
    if (tid == 0) {
      out_scores[row] = valid ? hyp_score[src] : -1000.0f;
      int L = 0;
      if (valid)
        for (int p = 0; p < SMAX; ++p) L += (hyp_tok[src][p] != PAD);
      out_lens[row] = L;
    }
  }
}

extern "C" void kernel_launch(void* const* d_in, const int* in_sizes, int n_in,
                              void* d_out, int out_size, void* d_ws, size_t ws_size,
                              hipStream_t stream) {
  (void)n_in; (void)out_size; (void)d_ws; (void)ws_size;
  const float* all_scores = (const float*)d_in[0];
  const int B = in_sizes[0] / (BEAM * TSEQ * VOCAB);   // 32

  int*   out_seqs   = (int*)d_out;                       // B*BEAM*SMAX int32 (bit-cast region)
  int*   out_lens   = out_seqs + B * BEAM * SMAX;        // B*BEAM int32
  float* out_scores = (float*)d_out + (size_t)B * BEAM * SMAX + B * BEAM;

  beam_search_kernel<<<B, BLOCK, 0, stream>>>(all_scores, out_seqs, out_lens, out_scores);
}